// MTSPSoftassign_9655086481584
// MI455X (gfx1250) — compile-verified
//
#include <hip/hip_runtime.h>
#include <cstdint>

#define BB   32
#define HH   8
#define NN   512
#define EPSV 1e-8f
#define NLAYERS 10
#define PIPE 3

typedef float v2f __attribute__((ext_vector_type(2)));
typedef float v8f __attribute__((ext_vector_type(8)));

// ---------------- helpers ----------------

__device__ __forceinline__ float wave_sum32(float v) {
#pragma unroll
  for (int off = 16; off > 0; off >>= 1) v += __shfl_down(v, off, 32);
  return v;
}

// CDNA5 async global->LDS copy (ASYNCcnt-tracked). VDST = LDS byte offset,
// VADDR = 64-bit global address, GV mode (no SADDR).
__device__ __forceinline__ void async_copy_b128(uint32_t lds_off, const void* g) {
  asm volatile("global_load_async_to_lds_b128 %0, %1, off"
               :: "v"(lds_off), "v"(g) : "memory");
}

__device__ __forceinline__ void wait_async(int wt) {
  switch (wt) {
    case 0:  asm volatile("s_wait_asynccnt 0x0" ::: "memory"); break;
    case 1:  asm volatile("s_wait_asynccnt 0x1" ::: "memory"); break;
    default: asm volatile("s_wait_asynccnt 0x2" ::: "memory"); break;
  }
}

// ---------------- stage 0: maxes ----------------

// depotMax[b*8+h] = max_j In[b,h,0,j]
__global__ void depot_max_kernel(const float* __restrict__ in, float* __restrict__ depotMax) {
  int bh = blockIdx.x;                       // 0..255
  const float* row = in + (size_t)bh * (NN * NN);
  int t = threadIdx.x;                       // 256 threads
  __shared__ float sm[256];
  sm[t] = fmaxf(row[t], row[t + 256]);
  __syncthreads();
  for (int s = 128; s > 0; s >>= 1) {
    if (t < s) sm[t] = fmaxf(sm[t], sm[t + s]);
    __syncthreads();
  }
  if (t == 0) depotMax[bh] = sm[0];
}

// othersMax[b*512+i] = max_{h,j} In[b,h,i,j]   (i >= 1)
__global__ void others_max_kernel(const float* __restrict__ in, float* __restrict__ othersMax) {
  int b = blockIdx.x >> 9;
  int i = blockIdx.x & 511;
  if (i == 0) return;
  const float* base = in + (size_t)b * HH * NN * NN + (size_t)i * NN;
  int t = threadIdx.x;
  float m = -3.4e38f;
  for (int h = 0; h < HH; ++h) {
    const float* r = base + (size_t)h * NN * NN;
    m = fmaxf(m, fmaxf(r[t], r[t + 256]));
  }
  __shared__ float sm[256];
  sm[t] = m;
  __syncthreads();
  for (int s = 128; s > 0; s >>= 1) {
    if (t < s) sm[t] = fmaxf(sm[t], sm[t + s]);
    __syncthreads();
  }
  if (t == 0) othersMax[b * 512 + i] = sm[0];
}

// ---------------- stage 1: exp + layer-0 row partials ----------------
// grid = B*H*64 blocks, 256 threads; each wave handles one full row (512 cols).
// partial layout (row mode): partial[(b*8 + h)*512 + i] = sum_j max(Y, EPS)
__global__ void exp_row_kernel(const float* __restrict__ in, float* __restrict__ out,
                               const float* __restrict__ depotMax,
                               const float* __restrict__ othersMax,
                               float* __restrict__ partial) {
  int blk = blockIdx.x;
  int rowBlk = blk & 63;
  int bh = blk >> 6;
  int b = bh >> 3, h = bh & 7;
  int t = threadIdx.x, w = t >> 5, l = t & 31;
  int i = rowBlk * 8 + w;
  const float4* src = (const float4*)(in + (size_t)bh * NN * NN + (size_t)i * NN);
  float4* dst = (float4*)(out + (size_t)bh * NN * NN + (size_t)i * NN);
  float m = (i == 0) ? depotMax[bh] : othersMax[b * 512 + i];
  float acc = 0.f;
#pragma unroll
  for (int k = 0; k < 4; ++k) {
    float4 v = src[l + 32 * k];
    float4 e;
    e.x = expf(v.x - m); e.y = expf(v.y - m);
    e.z = expf(v.z - m); e.w = expf(v.w - m);
    dst[l + 32 * k] = e;
    acc += fmaxf(e.x, EPSV) + fmaxf(e.y, EPSV) + fmaxf(e.z, EPSV) + fmaxf(e.w, EPSV);
  }
  acc = wave_sum32(acc);
  if (l == 0) partial[((size_t)b * HH + h) * 512 + i] = acc;
}

// ---------------- partial -> sums reduction (uses V_WMMA_F32_16X16X4_F32) ---
// partial: [b][nPairs][512]. othersSum[b][r] = sum_p partial[b][p][r].
// A = all-ones 16x4, B[k][n] = partial rows -> D[m][n] = column sums.
// depot: nPairs==8  -> depotSum[b][h] = partial[b][h][0]
//        nPairs==32 -> depotSum[b][h] = sum_c partial[b][h*4+c][0]
__global__ void reduce_sum_kernel(const float* __restrict__ partial,
                                  float* __restrict__ depotSum,
                                  float* __restrict__ othersSum, int nPairs) {
  int b = blockIdx.x >> 5;
  int g16 = blockIdx.x & 31;            // 16-column group
  int lane = threadIdx.x;               // 32 threads = 1 wave, EXEC all ones
  int n = lane & 15, hi = lane >> 4;
  const float* P = partial + (size_t)b * nPairs * 512;
  int col = g16 * 16 + n;
  v8f c = {};
  v2f a; a.x = 1.0f; a.y = 1.0f;
  for (int g = 0; g < nPairs; g += 4) {
    v2f bm;
    bm.x = P[(size_t)(g + 0 + 2 * hi) * 512 + col];   // K = 0 / 2
    bm.y = P[(size_t)(g + 1 + 2 * hi) * 512 + col];   // K = 1 / 3
    c = __builtin_amdgcn_wmma_f32_16x16x4_f32(false, a, false, bm,
                                              (short)0, c, false, false);
  }
  if (lane < 16) othersSum[b * 512 + col] = c[0];     // D[0][n] = column sum
  if (g16 == 0 && lane < 8) {
    float d;
    if (nPairs == 8) {
      d = P[(size_t)lane * 512];
    } else {
      d = P[(size_t)(lane * 4 + 0) * 512] + P[(size_t)(lane * 4 + 1) * 512] +
          P[(size_t)(lane * 4 + 2) * 512] + P[(size_t)(lane * 4 + 3) * 512];
    }
    depotSum[b * 8 + lane] = d;
  }
}

// ---------------- parity-0 apply: row normalize, emit column partials -------
// grid = B*H*4 (128-row chunks), 256 threads. 2-row tiles streamed through a
// 3-deep async-to-LDS pipeline; thread t owns cols 4*(t&127).. of row (t>>7).
// partial layout (col mode): partial[(b*32 + h*4 + chunk)*512 + j]
__global__ void apply_row_kernel(float* __restrict__ data,
                                 const float* __restrict__ depotSum,
                                 const float* __restrict__ othersSum,
                                 float* __restrict__ partial, int writePartial) {
  __shared__ float tile[PIPE * 1024];
  __shared__ float sinv[128];
  __shared__ float comb[512];
  int blk = blockIdx.x;
  int chunk = blk & 3;
  int bh = blk >> 2;
  int b = bh >> 3, h = bh & 7;
  int t = threadIdx.x;
  int cc = t & 127, rsel = t >> 7;
  float* gbase = data + (size_t)bh * NN * NN + (size_t)(chunk * 128) * NN;

  if (t < 128) {
    int r = chunk * 128 + t;
    float s = (r == 0) ? depotSum[bh] : othersSum[b * 512 + r];
    sinv[t] = 1.0f / s;
  }
  __syncthreads();

  uint32_t ldsBase = (uint32_t)(uintptr_t)(&tile[0]);
  uint32_t myOff = ldsBase + (uint32_t)(t * 16);
#pragma unroll
  for (int k = 0; k < PIPE; ++k) {
    const float* g = gbase + (size_t)(2 * k + rsel) * NN + cc * 4;
    async_copy_b128(myOff + (uint32_t)(k * 4096), g);
  }

  float a0 = 0.f, a1 = 0.f, a2 = 0.f, a3 = 0.f;
  for (int k = 0; k < 64; ++k) {
    int wt = 63 - k; if (wt > PIPE - 1) wt = PIPE - 1;
    wait_async(wt);
    int buf = k % PIPE;
    float4 v = *(const float4*)&tile[buf * 1024 + t * 4];
    float inv = sinv[2 * k + rsel];
    float4 o;
    o.x = fmaxf(v.x, EPSV) * inv; o.y = fmaxf(v.y, EPSV) * inv;
    o.z = fmaxf(v.z, EPSV) * inv; o.w = fmaxf(v.w, EPSV) * inv;
    *(float4*)(gbase + (size_t)(2 * k + rsel) * NN + cc * 4) = o;
    a0 += fmaxf(o.x, EPSV); a1 += fmaxf(o.y, EPSV);
    a2 += fmaxf(o.z, EPSV); a3 += fmaxf(o.w, EPSV);
    int kn = k + PIPE;
    if (kn < 64) {
      const float* g = gbase + (size_t)(2 * kn + rsel) * NN + cc * 4;
      async_copy_b128(myOff + (uint32_t)(buf * 4096), g);
    }
  }

  if (writePartial) {
    if (rsel == 1) { float4 st = {a0, a1, a2, a3}; *(float4*)&comb[cc * 4] = st; }
    __syncthreads();
    if (rsel == 0) {
      float4 o = *(const float4*)&comb[cc * 4];
      float4 res = {a0 + o.x, a1 + o.y, a2 + o.z, a3 + o.w};
      *(float4*)&partial[((size_t)b * 32 + h * 4 + chunk) * 512 + cc * 4] = res;
    }
  }
}

// ---------------- parity-1 apply: column normalize, emit row partials -------
// grid = B*H*64 (8-row blocks), 256 threads, wave-per-row.
__global__ void apply_col_kernel(float* __restrict__ data,
                                 const float* __restrict__ depotSum,
                                 const float* __restrict__ othersSum,
                                 float* __restrict__ partial, int writePartial) {
  __shared__ float cinv[512];
  int blk = blockIdx.x;
  int rowBlk = blk & 63;
  int bh = blk >> 6;
  int b = bh >> 3, h = bh & 7;
  int t = threadIdx.x, w = t >> 5, l = t & 31;
  for (int j = t; j < 512; j += 256) {
    float s = (j == 0) ? depotSum[bh] : othersSum[b * 512 + j];
    cinv[j] = 1.0f / s;
  }
  __syncthreads();
  int i = rowBlk * 8 + w;
  float4* rowp = (float4*)(data + (size_t)bh * NN * NN + (size_t)i * NN);
  float acc = 0.f;
#pragma unroll
  for (int k = 0; k < 4; ++k) {
    int c4 = l + 32 * k;
    float4 v = rowp[c4];
    float4 s = *(const float4*)&cinv[c4 * 4];
    float4 o;
    o.x = fmaxf(v.x, EPSV) * s.x; o.y = fmaxf(v.y, EPSV) * s.y;
    o.z = fmaxf(v.z, EPSV) * s.z; o.w = fmaxf(v.w, EPSV) * s.w;
    rowp[c4] = o;
    acc += fmaxf(o.x, EPSV) + fmaxf(o.y, EPSV) + fmaxf(o.z, EPSV) + fmaxf(o.w, EPSV);
  }
  acc = wave_sum32(acc);
  if (writePartial && l == 0) partial[((size_t)b * HH + h) * 512 + i] = acc;
}

// ---------------- launch ----------------

extern "C" void kernel_launch(void* const* d_in, const int* in_sizes, int n_in,
                              void* d_out, int out_size, void* d_ws, size_t ws_size,
                              hipStream_t stream) {
  const float* in = (const float*)d_in[0];
  float* out = (float*)d_out;
  float* ws = (float*)d_ws;

  float* depotMax  = ws;              // 256
  float* othersMax = ws + 256;        // 16384
  float* depotSum  = ws + 16640;      // 256
  float* othersSum = ws + 16896;      // 16384
  float* partial   = ws + 33280;      // 32*32*512 = 524288  (~2.2 MB total)

  depot_max_kernel <<<BB * HH, 256, 0, stream>>>(in, depotMax);
  others_max_kernel<<<BB * 512, 256, 0, stream>>>(in, othersMax);
  exp_row_kernel   <<<BB * HH * 64, 256, 0, stream>>>(in, out, depotMax, othersMax, partial);
  reduce_sum_kernel<<<BB * 32, 32, 0, stream>>>(partial, depotSum, othersSum, 8);

  for (int layer = 0; layer < NLAYERS; ++layer) {
    int parity = layer & 1;
    int last = (layer == NLAYERS - 1);
    if (parity == 0) {
      apply_row_kernel<<<BB * HH * 4, 256, 0, stream>>>(out, depotSum, othersSum, partial, !last);
      if (!last) reduce_sum_kernel<<<BB * 32, 32, 0, stream>>>(partial, depotSum, othersSum, 32);
    } else {
      apply_col_kernel<<<BB * HH * 64, 256, 0, stream>>>(out, depotSum, othersSum, partial, !last);
      if (!last) reduce_sum_kernel<<<BB * 32, 32, 0, stream>>>(partial, depotSum, othersSum, 8);
    }
  }
}